// DomainDiscriminator_49615462203904
// MI455X (gfx1250) — compile-verified
//
#include <hip/hip_runtime.h>
#include <math.h>

// ---- problem constants (match reference) ----
#define BATCH   64
#define NPG     400
#define TOPK    250
#define FEAT    64
#define SEQ     12
#define LIN     (TOPK * FEAT)   // 16000
#define HID     8000
#define NNODES  (BATCH * NPG)   // 25600
#define BN_EPS  1e-5f

typedef __attribute__((ext_vector_type(2))) float f32x2;
typedef __attribute__((ext_vector_type(8))) float f32x8;

// ---------------------------------------------------------------------------
// Kernel 1: per-graph score -> exact top-K (stable, descending) -> gather,
// tanh rescale, BatchNorm (eval). One block per graph.
// ---------------------------------------------------------------------------
__global__ __launch_bounds__(512) void pool_bn_kernel(
    const float* __restrict__ features,   // [SEQ, NNODES, FEAT]
    const float* __restrict__ w,          // [FEAT]
    const float* __restrict__ bn_gamma,   // [LIN]
    const float* __restrict__ bn_beta,    // [LIN]
    const float* __restrict__ bn_mean,    // [LIN]
    const float* __restrict__ bn_var,     // [LIN]
    float* __restrict__ xn)               // [BATCH, LIN]
{
    __shared__ float s_score[NPG];
    __shared__ float s_tanh[TOPK];
    __shared__ int   s_src[TOPK];
    __shared__ float s_invn;

    const int b   = blockIdx.x;
    const int tid = threadIdx.x;
    // x = features[-1], rows of this graph
    const float* __restrict__ x =
        features + (size_t)(SEQ - 1) * NNODES * FEAT + (size_t)b * NPG * FEAT;

    if (tid == 0) {
        float s = 0.f;
        for (int f = 0; f < FEAT; ++f) s += w[f] * w[f];
        s_invn = 1.0f / sqrtf(s);
    }
    __syncthreads();
    const float invn = s_invn;

    // scores
    for (int i = tid; i < NPG; i += 512) {
        const float* row = x + i * FEAT;
        float acc = 0.f;
        #pragma unroll 8
        for (int f = 0; f < FEAT; ++f) acc += row[f] * w[f];
        s_score[i] = acc * invn;
    }
    __syncthreads();

    // exact rank (stable tie-break = lower index first, like lax.top_k)
    for (int i = tid; i < NPG; i += 512) {
        const float si = s_score[i];
        int rank = 0;
        for (int j = 0; j < NPG; ++j) {
            const float sj = s_score[j];
            rank += (sj > si) || (sj == si && j < i);
        }
        if (rank < TOPK) { s_src[rank] = i; s_tanh[rank] = tanhf(si); }
    }
    __syncthreads();

    // coalesced gather + tanh scale + BatchNorm(eval)
    const size_t obase = (size_t)b * LIN;
    for (int idx = tid; idx < LIN; idx += 512) {
        const int r = idx >> 6;        // /FEAT
        const int f = idx & 63;        // %FEAT
        const float v  = x[s_src[r] * FEAT + f] * s_tanh[r];
        const float nv = (v - bn_mean[idx]) * rsqrtf(bn_var[idx] + BN_EPS)
                         * bn_gamma[idx] + bn_beta[idx];
        xn[obase + idx] = nv;
    }
}

// ---------------------------------------------------------------------------
// Kernel 2: h = relu(xn @ W1^T + b1)   [64,16000] x [8000,16000]^T -> [64,8000]
// One block (4 waves) per 16-wide N tile; each wave covers all 4 M-tiles and
// 1/4 of K with V_WMMA_F32_16X16X4_F32; LDS reduction + bias + ReLU.
// W1 is streamed once (non-temporal + prefetch); xn stays hot in L2.
// ---------------------------------------------------------------------------
__global__ __launch_bounds__(128) void gemm1_wmma_kernel(
    const float* __restrict__ xn,   // [64, LIN]
    const float* __restrict__ W1,   // [HID, LIN]
    const float* __restrict__ b1,   // [HID]
    float* __restrict__ h)          // [64, HID]
{
    __shared__ float s_part[4][64 * 16];   // 16 KB

    const int nb   = blockIdx.x;           // 0..499
    const int wv   = threadIdx.x >> 5;     // wave id 0..3 -> K split
    const int lane = threadIdx.x & 31;
    const int lrow = lane & 15;            // local M (for A) / local N (for B)
    const int lgrp = lane >> 4;            // 0/1 -> K sub-offset 2*lgrp

    const int n = nb * 16 + lrow;
    const float* __restrict__ wrow = W1 + (size_t)n * LIN;
    const float* __restrict__ a0 = xn + (size_t)( 0 + lrow) * LIN;
    const float* __restrict__ a1 = xn + (size_t)(16 + lrow) * LIN;
    const float* __restrict__ a2 = xn + (size_t)(32 + lrow) * LIN;
    const float* __restrict__ a3 = xn + (size_t)(48 + lrow) * LIN;

    const int k0 = wv * (LIN / 4);
    const int k1 = k0 + (LIN / 4);

    f32x8 c0 = {}, c1 = {}, c2 = {}, c3 = {};

    #pragma unroll 4
    for (int kb = k0; kb < k1; kb += 4) {
        const int kk = kb + 2 * lgrp;      // even -> 8B aligned f32x2 loads
        // B fragment: W1 streamed once -> non-temporal; prefetch 1KB ahead
        f32x2 bf  = __builtin_nontemporal_load((const f32x2*)(wrow + kk));
        __builtin_prefetch(wrow + kk + 256, 0, 1);
        // A fragments (L2-resident activations)
        f32x2 af0 = *(const f32x2*)(a0 + kk);
        f32x2 af1 = *(const f32x2*)(a1 + kk);
        f32x2 af2 = *(const f32x2*)(a2 + kk);
        f32x2 af3 = *(const f32x2*)(a3 + kk);

        c0 = __builtin_amdgcn_wmma_f32_16x16x4_f32(false, af0, false, bf,
                                                   (short)0, c0, false, false);
        c1 = __builtin_amdgcn_wmma_f32_16x16x4_f32(false, af1, false, bf,
                                                   (short)0, c1, false, false);
        c2 = __builtin_amdgcn_wmma_f32_16x16x4_f32(false, af2, false, bf,
                                                   (short)0, c2, false, false);
        c3 = __builtin_amdgcn_wmma_f32_16x16x4_f32(false, af3, false, bf,
                                                   (short)0, c3, false, false);
    }

    // D layout: vgpr i, lane L -> (M = i + 8*(L/16), N = L%16)
    float* sp = s_part[wv];
    #pragma unroll
    for (int i = 0; i < 8; ++i) {
        const int m0 = i + 8 * lgrp;
        sp[( 0 + m0) * 16 + lrow] = c0[i];
        sp[(16 + m0) * 16 + lrow] = c1[i];
        sp[(32 + m0) * 16 + lrow] = c2[i];
        sp[(48 + m0) * 16 + lrow] = c3[i];
    }
    __syncthreads();

    for (int idx = threadIdx.x; idx < 64 * 16; idx += 128) {
        float s = s_part[0][idx] + s_part[1][idx] + s_part[2][idx] + s_part[3][idx];
        const int m   = idx >> 4;
        const int col = idx & 15;
        const int nn  = nb * 16 + col;
        s += b1[nn];
        h[(size_t)m * HID + nn] = fmaxf(s, 0.f);
    }
}

// ---------------------------------------------------------------------------
// Kernel 3: logits[b] = h[b,:] . W2 + b2
// ---------------------------------------------------------------------------
__global__ __launch_bounds__(256) void gemm2_kernel(
    const float* __restrict__ h, const float* __restrict__ W2,
    const float* __restrict__ b2, float* __restrict__ logits)
{
    __shared__ float red[256];
    const int b = blockIdx.x;
    float acc = 0.f;
    for (int j = threadIdx.x; j < HID; j += 256)
        acc += h[(size_t)b * HID + j] * W2[j];
    red[threadIdx.x] = acc;
    __syncthreads();
    for (int s = 128; s > 0; s >>= 1) {
        if (threadIdx.x < s) red[threadIdx.x] += red[threadIdx.x + s];
        __syncthreads();
    }
    if (threadIdx.x == 0) logits[b] = red[0] + b2[0];
}

// ---------------------------------------------------------------------------
// Kernel 4: loss = mean(softplus(x) - x)   (truth == 1), numerically stable
// ---------------------------------------------------------------------------
__global__ __launch_bounds__(64) void loss_kernel(
    const float* __restrict__ logits, float* __restrict__ out)
{
    __shared__ float red[64];
    const int t = threadIdx.x;
    const float x = logits[t];
    red[t] = fmaxf(x, 0.f) + log1pf(expf(-fabsf(x))) - x;
    __syncthreads();
    for (int s = 32; s > 0; s >>= 1) {
        if (t < s) red[t] += red[t + s];
        __syncthreads();
    }
    if (t == 0) out[0] = red[0] * (1.0f / (float)BATCH);
}

// ---------------------------------------------------------------------------
extern "C" void kernel_launch(void* const* d_in, const int* in_sizes, int n_in,
                              void* d_out, int out_size, void* d_ws, size_t ws_size,
                              hipStream_t stream) {
    const float* features = (const float*)d_in[0];
    // d_in[1] edge_index (int64, unused), d_in[2] batch (int64, unused)
    const float* w        = (const float*)d_in[3];
    const float* bn_gamma = (const float*)d_in[4];
    const float* bn_beta  = (const float*)d_in[5];
    const float* bn_mean  = (const float*)d_in[6];
    const float* bn_var   = (const float*)d_in[7];
    const float* W1       = (const float*)d_in[8];
    const float* b1       = (const float*)d_in[9];
    const float* W2       = (const float*)d_in[10];
    const float* b2       = (const float*)d_in[11];
    float* out            = (float*)d_out;

    char* ws = (char*)d_ws;
    float* xn     = (float*)ws;                                           // 64*16000 f32
    float* h      = (float*)(ws + (size_t)BATCH * LIN * sizeof(float));   // 64*8000 f32
    float* logits = (float*)(ws + (size_t)BATCH * LIN * sizeof(float)
                                + (size_t)BATCH * HID * sizeof(float));   // 64 f32

    pool_bn_kernel<<<BATCH, 512, 0, stream>>>(features, w, bn_gamma, bn_beta,
                                              bn_mean, bn_var, xn);
    gemm1_wmma_kernel<<<HID / 16, 128, 0, stream>>>(xn, W1, b1, h);
    gemm2_kernel<<<BATCH, 256, 0, stream>>>(h, W2, b2, logits);
    loss_kernel<<<1, 64, 0, stream>>>(logits, out);
}